// SplineLoss_48945447305361
// MI455X (gfx1250) — compile-verified
//
#include <hip/hip_runtime.h>

typedef __attribute__((ext_vector_type(2))) float v2f;
typedef __attribute__((ext_vector_type(8))) float v8f;

namespace {
constexpr int   kB       = 1024;      // rows per input
constexpr int   kN       = 8192;      // columns per input
constexpr int   kStride  = 10;        // FRAME_INTERVAL
constexpr int   kSamples = 820;       // len(range(0, 8192, 10))
constexpr float kTotal   = 839680.0f; // 820 * 1024 (mean divisor)
constexpr int   kThreads = 256;       // 8 waves (wave32)
}

// Full-wave (32 lane) sum via V_WMMA_F32_16X16X4_F32.
// A (16x4 f32) layout: lanes 0-15 VGPR0 = (M=lane, K=0), VGPR1 = K=1;
// lanes 16-31 VGPR0 = (M=lane-16, K=2), VGPR1 = K=3.  With a.y = 0 and
// B = all-ones, D[m][n] = s[m] + s[m+16] for every n.  D register v holds
// row v (lanes 0-15) and row v+8 (lanes 16-31), so sum(c[0..7]) gives rows
// 0-7+16-23 on the low half-wave and rows 8-15+24-31 on the high half-wave;
// one cross-half shuffle completes the 32-lane sum (valid in lanes 0-15 only).
__device__ __forceinline__ float wave_reduce_wmma(float v) {
  v2f a;    a.x = v;       a.y = 0.0f;
  v2f ones; ones.x = 1.0f; ones.y = 1.0f;
  v8f c = {};
  c = __builtin_amdgcn_wmma_f32_16x16x4_f32(
      /*neg_a=*/false, a, /*neg_b=*/false, ones,
      /*c_mod=*/(short)0, c, /*reuse_a=*/false, /*reuse_b=*/false);
  float part = ((c[0] + c[1]) + (c[2] + c[3])) + ((c[4] + c[5]) + (c[6] + c[7]));
  return part + __shfl(part, 16, 32);
}

// Deterministic block reduction: per-wave WMMA reduce (all waves fully
// active -> EXEC all-1s at the WMMA), then 8 partials folded by thread 0.
__device__ __forceinline__ float block_reduce(float v, float* red) {
  const int tid  = threadIdx.x;
  const int lane = tid & 31;
  const int wave = tid >> 5;
  const float wsum = wave_reduce_wmma(v);
  if (lane == 0) red[wave] = wsum;
  __syncthreads();
  float tot = 0.0f;
  if (tid == 0) {
#pragma unroll
    for (int i = 0; i < kThreads / 32; ++i) tot += red[i];
  }
  return tot;  // valid on thread 0 only
}

// Kernel 1: one block per row b.  Since the integer sample times land on the
// knots (t == 0 in the spline eval), the reference collapses to a strided
// squared-difference sum:  sum_k (true[b,10k] - pred[b,10k])^2.
__global__ void __launch_bounds__(kThreads)
spline_mse_partial(const float* __restrict__ tf,
                   const float* __restrict__ pf,
                   float* __restrict__ partial) {
  const int b = blockIdx.x;
  const float* __restrict__ trow = tf + (size_t)b * kN;
  const float* __restrict__ prow = pf + (size_t)b * kN;
  float acc = 0.0f;
  for (int k = threadIdx.x; k < kSamples; k += kThreads) {
    const int n = k * kStride;
    const float d = __builtin_nontemporal_load(trow + n) -
                    __builtin_nontemporal_load(prow + n);
    acc += d * d;
  }
  __shared__ float red[kThreads / 32];
  const float tot = block_reduce(acc, red);
  if (threadIdx.x == 0) partial[b] = tot;
}

// Kernel 2: single block folds the 1024 row partials and writes the mean.
__global__ void __launch_bounds__(kThreads)
spline_mse_final(const float* __restrict__ partial, float* __restrict__ out) {
  float acc = 0.0f;
#pragma unroll
  for (int i = threadIdx.x; i < kB; i += kThreads) acc += partial[i];
  __shared__ float red[kThreads / 32];
  const float tot = block_reduce(acc, red);
  if (threadIdx.x == 0) out[0] = tot / kTotal;
}

extern "C" void kernel_launch(void* const* d_in, const int* in_sizes, int n_in,
                              void* d_out, int out_size, void* d_ws, size_t ws_size,
                              hipStream_t stream) {
  const float* tf = (const float*)d_in[0];   // true_frames  (1024 x 8192 f32)
  const float* pf = (const float*)d_in[1];   // predicted_frames
  float* partial  = (float*)d_ws;            // kB floats of scratch (written before read)
  float* out      = (float*)d_out;           // single f32 (overwritten every call)

  spline_mse_partial<<<kB, kThreads, 0, stream>>>(tf, pf, partial);
  spline_mse_final<<<1, kThreads, 0, stream>>>(partial, out);
}